// GraphConvolution_53197464929045
// MI455X (gfx1250) — compile-verified
//
#include <hip/hip_runtime.h>
#include <hip/hip_bf16.h>

typedef float v2f __attribute__((ext_vector_type(2)));
typedef float v8f __attribute__((ext_vector_type(8)));

// ---------------------------------------------------------------------------
// 1) degree accumulation: deg_out[src[e]] += 1, deg_in[dst[e]] += 1
// ---------------------------------------------------------------------------
__global__ __launch_bounds__(256)
void gcn_degree_kernel(const long long* __restrict__ src,
                       const long long* __restrict__ dst,
                       float* __restrict__ degOut, float* __restrict__ degIn,
                       int E) {
    int i = blockIdx.x * blockDim.x + threadIdx.x;
    if (i < E) {
        atomicAdd(&degOut[(int)src[i]], 1.0f);
        atomicAdd(&degIn [(int)dst[i]], 1.0f);
    }
}

// deg -> norm in place: deg>0 ? rsqrt(max(deg,1)) : 0
__global__ __launch_bounds__(256)
void gcn_norm_kernel(float* __restrict__ degOut, float* __restrict__ degIn, int n) {
    int i = blockIdx.x * blockDim.x + threadIdx.x;
    if (i < n) {
        float a = degOut[i];
        float b = degIn[i];
        degOut[i] = (a > 0.0f) ? rsqrtf(fmaxf(a, 1.0f)) : 0.0f;
        degIn[i]  = (b > 0.0f) ? rsqrtf(fmaxf(b, 1.0f)) : 0.0f;
    }
}

// ---------------------------------------------------------------------------
// 2) WMMA GEMM: OUT[16 rows x dOut] = (X * normS) @ W,  X:[n,128], W:[128,dOut]
//    block = 256 threads = 8 waves; block handles 16 rows; wave w handles
//    column tiles w, w+8, ... (dOut/16 tiles). K-loop: 32 x wmma 16x16x4 f32.
// ---------------------------------------------------------------------------
__global__ __launch_bounds__(256)
void gcn_gemm_wmma_kernel(const float* __restrict__ X,
                          const float* __restrict__ normS,
                          const float* __restrict__ W,
                          float* __restrict__ OUT,
                          int dOut) {
    __shared__ float As[16 * 128];            // 8 KB A tile (pre-scaled)

    const int tid     = threadIdx.x;
    const int rowBase = blockIdx.x * 16;

    // Stage A tile into LDS, scaling each row by norm_src. 2048 floats,
    // 256 threads -> 2 float4 each.
    for (int i = tid; i < 16 * 128 / 4; i += 256) {
        int r = (i * 4) / 128;
        int c = (i * 4) % 128;
        const float4 v = *reinterpret_cast<const float4*>(
            X + (size_t)(rowBase + r) * 128 + c);
        float nv = normS[rowBase + r];
        float4 o;
        o.x = v.x * nv; o.y = v.y * nv; o.z = v.z * nv; o.w = v.w * nv;
        *reinterpret_cast<float4*>(&As[r * 128 + c]) = o;
    }
    __syncthreads();

    const int wave = tid >> 5;                // 0..7
    const int lane = tid & 31;
    const int half = lane >> 4;               // 0 or 1 (K/row-half select)
    const int l16  = lane & 15;
    const int nColTiles = dOut >> 4;          // 8 (d=128) or 16 (d=256)

    for (int ct = wave; ct < nColTiles; ct += 8) {   // wave-uniform loop
        const int colBase = ct * 16;
        v8f acc = {};
        for (int k = 0; k < 128; k += 4) {
            const int kb = k + half * 2;
            v2f a, b;
            // A 16x4 f32 layout: lanes 0-15 rows 0-15 K={0,1}; lanes 16-31 K={2,3}
            a.x = As[l16 * 128 + kb];
            a.y = As[l16 * 128 + kb + 1];
            // B 4x16 f32 layout (mirror): lane = column, K-pair by lane half
            b.x = W[(size_t)kb       * dOut + colBase + l16];
            b.y = W[(size_t)(kb + 1) * dOut + colBase + l16];
            acc = __builtin_amdgcn_wmma_f32_16x16x4_f32(
                false, a, false, b, (short)0, acc, false, false);
        }
        // C/D 16x16: lane = col (l16), VGPR r = row r + half*8
        float* o = OUT + (size_t)(rowBase + half * 8) * dOut + colBase + l16;
#pragma unroll
        for (int r = 0; r < 8; ++r)
            o[(size_t)r * dOut] = acc[r];
    }
}

// ---------------------------------------------------------------------------
// 3) SpMM scatter: agg[dst[e], :] += XW[src[e], :]   (wave per edge)
// ---------------------------------------------------------------------------
__global__ __launch_bounds__(256)
void gcn_scatter_kernel(const float* __restrict__ XW,
                        const long long* __restrict__ src,
                        const long long* __restrict__ dst,
                        float* __restrict__ agg,
                        int E, int dOut) {
    const int gwave  = (blockIdx.x * blockDim.x + threadIdx.x) >> 5;
    const int lane   = threadIdx.x & 31;
    const int nWaves = (gridDim.x * blockDim.x) >> 5;
    for (int e = gwave; e < E; e += nWaves) {
        const int s = (int)src[e];
        const int d = (int)dst[e];
        const float* xp = XW  + (size_t)s * dOut;
        float*       ap = agg + (size_t)d * dOut;
        for (int c = lane; c < dOut; c += 32)
            atomicAdd(&ap[c], xp[c]);
    }
}

// ---------------------------------------------------------------------------
// 4) finalize: out = agg * norm_dst + b  (optional ReLU)
// ---------------------------------------------------------------------------
__global__ __launch_bounds__(256)
void gcn_finalize_kernel(const float* __restrict__ agg,
                         const float* __restrict__ normD,
                         const float* __restrict__ bias,
                         float* __restrict__ out,
                         int n, int dOut, int doRelu) {
    size_t idx   = (size_t)blockIdx.x * blockDim.x + threadIdx.x;
    size_t total = (size_t)n * dOut;
    if (idx < total) {
        int c = (int)(idx % dOut);
        int r = (int)(idx / dOut);
        float v = agg[idx] * normD[r] + bias[c];
        out[idx] = doRelu ? fmaxf(v, 0.0f) : v;
    }
}

// int64 edge index -> float (used only if harness expects converted values)
__global__ __launch_bounds__(256)
void gcn_ei_to_float_kernel(const long long* __restrict__ ei,
                            float* __restrict__ out, int n) {
    int i = blockIdx.x * blockDim.x + threadIdx.x;
    if (i < n) out[i] = (float)ei[i];
}

// ---------------------------------------------------------------------------
extern "C" void kernel_launch(void* const* d_in, const int* in_sizes, int n_in,
                              void* d_out, int out_size, void* d_ws, size_t ws_size,
                              hipStream_t stream) {
    (void)n_in; (void)ws_size;
    const float*     x   = (const float*)d_in[0];
    const long long* ei  = (const long long*)d_in[1];
    const float*     W1  = (const float*)d_in[2];
    const float*     b1  = (const float*)d_in[3];
    const float*     W2  = (const float*)d_in[4];
    const float*     b2  = (const float*)d_in[5];
    const float*     W3  = (const float*)d_in[6];
    const float*     b3  = (const float*)d_in[7];

    const int N   = in_sizes[0] / 128;       // 100000
    const int E   = in_sizes[1] / 2;         // 1600000
    const int dO3 = in_sizes[6] / 128;       // 256

    const long long* src = ei;
    const long long* dst = ei + E;

    // Workspace carve-up (all 16B-aligned by construction)
    char*  ws     = (char*)d_ws;
    float* normS  = (float*)ws;  ws += sizeof(float) * (size_t)N;
    float* normD  = (float*)ws;  ws += sizeof(float) * (size_t)N;
    float* bufXW  = (float*)ws;  ws += sizeof(float) * (size_t)N * 256;
    float* bufAgg = (float*)ws;  ws += sizeof(float) * (size_t)N * 256;
    float* bufH   = (float*)ws;  ws += sizeof(float) * (size_t)N * 128;

    // ---- degree norms (computed once, reused for all 3 layers) ----
    hipMemsetAsync(normS, 0, sizeof(float) * (size_t)N, stream);
    hipMemsetAsync(normD, 0, sizeof(float) * (size_t)N, stream);
    gcn_degree_kernel<<<(E + 255) / 256, 256, 0, stream>>>(src, dst, normS, normD, E);
    gcn_norm_kernel<<<(N + 255) / 256, 256, 0, stream>>>(normS, normD, N);

    const int rowTiles = N / 16;                       // N is a multiple of 16
    const int scatterBlocks = 8192;                    // 65536 waves grid-stride

    // ---- layer 1: 128 -> 128, ReLU ----
    gcn_gemm_wmma_kernel<<<rowTiles, 256, 0, stream>>>(x, normS, W1, bufXW, 128);
    hipMemsetAsync(bufAgg, 0, sizeof(float) * (size_t)N * 128, stream);
    gcn_scatter_kernel<<<scatterBlocks, 256, 0, stream>>>(bufXW, src, dst, bufAgg, E, 128);
    gcn_finalize_kernel<<<(int)(((size_t)N * 128 + 255) / 256), 256, 0, stream>>>(
        bufAgg, normD, b1, bufH, N, 128, 1);

    // ---- layer 2: 128 -> 128, ReLU ----
    gcn_gemm_wmma_kernel<<<rowTiles, 256, 0, stream>>>(bufH, normS, W2, bufXW, 128);
    hipMemsetAsync(bufAgg, 0, sizeof(float) * (size_t)N * 128, stream);
    gcn_scatter_kernel<<<scatterBlocks, 256, 0, stream>>>(bufXW, src, dst, bufAgg, E, 128);
    gcn_finalize_kernel<<<(int)(((size_t)N * 128 + 255) / 256), 256, 0, stream>>>(
        bufAgg, normD, b2, bufH, N, 128, 1);

    // ---- layer 3: 128 -> 256, no ReLU, straight into d_out ----
    float* outF = (float*)d_out;
    gcn_gemm_wmma_kernel<<<rowTiles, 256, 0, stream>>>(bufH, normS, W3, bufXW, dO3);
    hipMemsetAsync(bufAgg, 0, sizeof(float) * (size_t)N * dO3, stream);
    gcn_scatter_kernel<<<scatterBlocks, 256, 0, stream>>>(bufXW, src, dst, bufAgg, E, dO3);
    gcn_finalize_kernel<<<(int)(((size_t)N * dO3 + 255) / 256), 256, 0, stream>>>(
        bufAgg, normD, b3, outF, N, dO3, 0);

    // ---- edge_index tail of the output tuple ----
    const size_t hElems = (size_t)N * dO3;             // 25.6M floats
    const size_t eiElems = (size_t)2 * E;              // 3.2M int64 values
    if ((size_t)out_size >= hElems + 2 * eiElems) {
        // raw int64 bytes reinterpreted as 2 floats each
        hipMemcpyAsync(outF + hElems, ei, eiElems * sizeof(long long),
                       hipMemcpyDeviceToDevice, stream);
    } else if ((size_t)out_size >= hElems + eiElems) {
        // values converted to float
        gcn_ei_to_float_kernel<<<(int)((eiElems + 255) / 256), 256, 0, stream>>>(
            ei, outF + hElems, (int)eiElems);
    }
}